// GaussianLoss_37357625541447
// MI455X (gfx1250) — compile-verified
//
#include <hip/hip_runtime.h>
#include <math.h>

typedef __attribute__((ext_vector_type(2))) float v2f;
typedef __attribute__((ext_vector_type(8))) float v8f;

#define B_  8
#define K_  16
#define NG  16            // gaussians per keypoint (N2/2)
#define H_  128
#define W_  128
#define HW_ (H_ * W_)
#define CHUNKS 8
#define PIX_PER_BLOCK (HW_ / CHUNKS)   // 2048
#define THREADS 256
#define WAVES (THREADS / 32)

#define LOG2E_F 1.4426950408889634f
#define LN2_F   0.6931471805599453f
#define SWAPX16 0x401f                 // ds_swizzle imm: xor_mask=0x10, and=0x1f

// Bare v_exp_f32 / v_log_f32 (args are range-safe by construction; deep
// underflow to 0/denorm matches f32 exp semantics without libm guard code).
__device__ __forceinline__ float fexp2(float x) { return __builtin_amdgcn_exp2f(x); }
__device__ __forceinline__ float flog2(float x) { return __builtin_amdgcn_logf(x); }

__device__ __forceinline__ float xhalf_swap(float x) {
  return __int_as_float(__builtin_amdgcn_ds_swizzle(__float_as_int(x), SWAPX16));
}

__device__ __forceinline__ float sum8_exp2(v8f d) {
  return ((fexp2(d[0]) + fexp2(d[1])) + (fexp2(d[2]) + fexp2(d[3]))) +
         ((fexp2(d[4]) + fexp2(d[5])) + (fexp2(d[6]) + fexp2(d[7])));
}

__global__ __launch_bounds__(THREADS)
void gauss_loss_partial(const float* __restrict__ centers,
                        const float* __restrict__ radius,
                        const float* __restrict__ mask,
                        const int*   __restrict__ ind,
                        const float* __restrict__ target,
                        const float* __restrict__ peak,
                        float* __restrict__ partial)
{
  __shared__ float s_cx[NG], s_cy[NG];
  __shared__ float s_scale, s_mask;
  __shared__ float s_red[WAVES];

  const int blk   = blockIdx.x;
  const int bk    = blk / CHUNKS;      // (b*K + k), 0..127
  const int chunk = blk % CHUNKS;
  const int b     = bk / K_;
  const int tid   = threadIdx.x;

  // ---- per-(b,k) gather: 16 centers (x,y) + peak, radius -> scale, mask ----
  if (tid < NG) {
    int pos  = ind[bk];
    float cx = centers[(size_t)(b * 32 + 2 * tid)     * HW_ + pos] + peak[bk * 2 + 0];
    float cy = centers[(size_t)(b * 32 + 2 * tid + 1) * HW_ + pos] + peak[bk * 2 + 1];
    s_cx[tid] = cx;
    s_cy[tid] = cy;
  } else if (tid == NG) {
    int pos = ind[bk];
    float r = radius[(size_t)b * HW_ + pos];
    // exp(-d2/(2 r^2)) == exp2(d2 * s),  s = -log2(e)/(2 r^2), folded into A
    s_scale = -LOG2E_F / (2.0f * r * r);
  } else if (tid == NG + 1) {
    s_mask = mask[bk];
  }
  __syncthreads();

  const int lane = tid & 31;
  const int wid  = tid >> 5;
  const int nl   = lane & 15;
  const bool lo_half = (lane < 16);
  const float s  = s_scale;
  const float m  = s_mask;

  // A-matrix (16 gaussians x 4), loop-invariant. Row n = s*[-2cx, -2cy, 1, cx^2+cy^2]
  // Lanes 0-15: VGPR0=K0, VGPR1=K1 ; lanes 16-31: VGPR0=K2, VGPR1=K3 (M = lane%16)
  v2f a;
  {
    float cx = s_cx[nl], cy = s_cy[nl];
    if (lo_half) { a.x = -2.0f * s * cx;  a.y = -2.0f * s * cy; }
    else         { a.x = s;               a.y = s * (cx * cx + cy * cy); }
  }

  const float* tgt = target + (size_t)bk * HW_;
  const int chunkBase = chunk * PIX_PER_BLOCK;
  const int perWave   = PIX_PER_BLOCK / WAVES;   // 256 pixels per wave
  const int iters     = perWave / 32;            // 8 iters x 32 pixels

  float acc = 0.0f;
#pragma unroll 2
  for (int it = 0; it < iters; ++it) {
    const int pbase = chunkBase + wid * perWave + it * 32;
    // B-matrix (4 x 16 pixels). Column p = [x, y, x^2+y^2, 1].
    // lo covers pixels pbase..+15, hi covers pbase+16..+31.
    const int qlo = pbase + nl;
    const int qhi = qlo + 16;
    v2f blo, bhi;
    {
      float xl = (float)(qlo & (W_ - 1)), yl = (float)(qlo >> 7);
      float xh = (float)(qhi & (W_ - 1)), yh = (float)(qhi >> 7);
      if (lo_half) { blo.x = xl;                blo.y = yl;
                     bhi.x = xh;                bhi.y = yh; }
      else         { blo.x = xl * xl + yl * yl; blo.y = 1.0f;
                     bhi.x = xh * xh + yh * yh; bhi.y = 1.0f; }
    }
    v8f c0 = {};
    // D[m][n] = s * d2(gaussian m, pixel n)  (scale folded into A)
    v8f d0 = __builtin_amdgcn_wmma_f32_16x16x4_f32(false, a, false, blo,
                                                   (short)0, c0, false, false);
    v8f d1 = __builtin_amdgcn_wmma_f32_16x16x4_f32(false, a, false, bhi,
                                                   (short)0, c0, false, false);
    // In-lane sum over 8 gaussians; lanes l/l^16 hold the two halves of the
    // same pixel's sum. Each lane sends the half the OTHER side needs: one
    // ds_swizzle (SWAPX16) completes both pixel groups.
    float plo  = sum8_exp2(d0);
    float phi  = sum8_exp2(d1);
    float mine = lo_half ? plo : phi;
    float send = lo_half ? phi : plo;
    float g    = mine + xhalf_swap(send);   // pixel pbase+lane's full 16-sum

    float t = tgt[pbase + lane];

    // BCE via log-sigmoid identities (no div, no log1p):
    //   e = exp(-g); log(sig) = -log(1+e); log(1-sig) = -g - log(1+e)
    // 1+e in [1,2] -> bare v_log is accurate there.
    float e     = fexp2(-g * LOG2E_F);          // g in (0,16]: range-safe
    float ln1pe = flog2(1.0f + e) * LN2_F;
    float lx    = fmaxf(-ln1pe,      -100.0f);
    float l1x   = fmaxf(-g - ln1pe,  -100.0f);
    acc += m * (t * lx + (1.0f - t) * l1x);     // ==0 when mask==0 (as ref)
  }

  // wave reduce (wave32) then block reduce
  for (int off = 16; off > 0; off >>= 1)
    acc += __shfl_xor(acc, off, 32);
  if (lane == 0) s_red[wid] = acc;
  __syncthreads();
  if (tid == 0) {
    float tot = 0.0f;
    for (int i = 0; i < WAVES; ++i) tot += s_red[i];
    partial[blk] = tot;
  }
}

__global__ __launch_bounds__(256)
void reduce_final(const float* __restrict__ partial, float* __restrict__ out,
                  int n, float inv_count)
{
  __shared__ float sm[256];
  float v = 0.0f;
  for (int i = threadIdx.x; i < n; i += 256) v += partial[i];
  sm[threadIdx.x] = v;
  __syncthreads();
  for (int s = 128; s > 0; s >>= 1) {
    if (threadIdx.x < (unsigned)s) sm[threadIdx.x] += sm[threadIdx.x + s];
    __syncthreads();
  }
  if (threadIdx.x == 0) {
    float loss = -sm[0] * inv_count;
    out[0] = loss;
    out[1] = loss;
  }
}

extern "C" void kernel_launch(void* const* d_in, const int* in_sizes, int n_in,
                              void* d_out, int out_size, void* d_ws, size_t ws_size,
                              hipStream_t stream) {
  const float* centers = (const float*)d_in[0];
  const float* radius  = (const float*)d_in[1];
  const float* mask    = (const float*)d_in[2];
  const int*   ind     = (const int*)d_in[3];
  const float* target  = (const float*)d_in[4];
  const float* peak    = (const float*)d_in[5];
  float* out     = (float*)d_out;
  float* partial = (float*)d_ws;

  const int nblocks = B_ * K_ * CHUNKS;   // 1024 partials, rewritten every launch
  gauss_loss_partial<<<nblocks, THREADS, 0, stream>>>(centers, radius, mask, ind,
                                                      target, peak, partial);
  const float inv_count = 1.0f / (float)((size_t)B_ * K_ * HW_);
  reduce_final<<<1, 256, 0, stream>>>(partial, out, nblocks, inv_count);
}